// GCN_88313117540590
// MI455X (gfx1250) — compile-verified
//
#include <hip/hip_runtime.h>

typedef __attribute__((ext_vector_type(2))) float v2f;
typedef __attribute__((ext_vector_type(8))) float v8f;

#define LDK 132   // transposed-W LDS stride in floats: 132 % 64 == 4 -> 32 b64 frag loads hit all 64 banks once

// ---------------------------------------------------------------------------
// GEMM: Y[nrows x 128] = X[nrows x 128] @ W[128 x 128], fp32 via V_WMMA_F32_16X16X4_F32
// W staged TRANSPOSED in LDS so a B fragment (K, K+1 of one column) is one ds_load_b64
// ---------------------------------------------------------------------------
__global__ __launch_bounds__(256) void k_gemm128(const float* __restrict__ X,
                                                 const float* __restrict__ W,
                                                 float* __restrict__ Y,
                                                 int nrows)
{
    __shared__ float sWT[128 * LDK];   // sWT[n * LDK + k] = W[k][n]

    const int wave = threadIdx.x >> 5;
    const int lane = threadIdx.x & 31;
    const int m0   = blockIdx.x * 128 + wave * 16;

    const int hh = (lane >> 4) << 1;       // K sub-offset: 0 (lanes 0-15) or 2 (lanes 16-31)
    const int ln = lane & 15;

    int arow = m0 + ln;
    if (arow > nrows - 1) arow = nrows - 1;          // clamp tail loads (stores are guarded)
    const float* xrow = X + (size_t)arow * 128 + hh;

    // warm L0 with this wave's A panel while W is staged (global_prefetch_b8)
    __builtin_prefetch(xrow, 0, 3);
    __builtin_prefetch(xrow + 32, 0, 3);
    __builtin_prefetch(xrow + 64, 0, 3);
    __builtin_prefetch(xrow + 96, 0, 3);

    // cooperative transposed load of W (coalesced global float4 reads)
    {
        const float4* W4 = (const float4*)W;
        for (int i = threadIdx.x; i < 128 * 32; i += 256) {
            float4 v = W4[i];
            int k  = i >> 5;           // W row (K index)
            int c4 = (i & 31) << 2;    // W col base (N index)
            sWT[(c4 + 0) * LDK + k] = v.x;
            sWT[(c4 + 1) * LDK + k] = v.y;
            sWT[(c4 + 2) * LDK + k] = v.z;
            sWT[(c4 + 3) * LDK + k] = v.w;
        }
    }
    __syncthreads();

    if (m0 >= nrows) return;               // wave-uniform: EXEC stays all-ones for WMMA

    v8f acc[8];
#pragma unroll
    for (int i = 0; i < 8; ++i) acc[i] = (v8f){0.f,0.f,0.f,0.f,0.f,0.f,0.f,0.f};

    // per-lane column base for B fragments: column ln (+ n0*16), K offset hh
    const float* wbase = sWT + (size_t)ln * LDK + hh;

    for (int k0 = 0; k0 < 128; k0 += 4) {
        // A fragment: lane(half h, m=ln) holds X[m0+ln][k0+2h + {0,1}]
        v2f a = *(const v2f*)(xrow + k0);
#pragma unroll
        for (int n0 = 0; n0 < 8; ++n0) {
            // B fragment: contiguous pair sWT[(n0*16+ln)*LDK + k0+2h .. +1] -> one ds_load_b64
            v2f b = *(const v2f*)(wbase + n0 * 16 * LDK + k0);
            acc[n0] = __builtin_amdgcn_wmma_f32_16x16x4_f32(
                false, a, false, b, (short)0, acc[n0], false, false);
        }
    }

    // C/D layout: VGPR v, lane-half h -> row M = v + 8h, col N = n0*16 + ln
    const int rbase = m0 + ((lane >> 4) << 3);
    if (rbase + 7 < nrows) {
        // fast path: whole 8-row slab in range
#pragma unroll
        for (int v = 0; v < 8; ++v) {
            float* yr = Y + (size_t)(rbase + v) * 128;
#pragma unroll
            for (int n0 = 0; n0 < 8; ++n0)
                yr[n0 * 16 + ln] = acc[n0][v];
        }
    } else {
#pragma unroll
        for (int v = 0; v < 8; ++v) {
            int row = rbase + v;
            if (row < nrows) {
                float* yr = Y + (size_t)row * 128;
#pragma unroll
                for (int n0 = 0; n0 < 8; ++n0)
                    yr[n0 * 16 + ln] = acc[n0][v];
            }
        }
    }
}

// ---------------------------------------------------------------------------
// Degree / normalization helpers
// ---------------------------------------------------------------------------
__global__ void k_zero_i32(int* __restrict__ p, int n) {
    int i = blockIdx.x * 256 + threadIdx.x;
    if (i < n) p[i] = 0;
}

__global__ void k_zero_f32(float* __restrict__ p, int n) {
    int i = blockIdx.x * 256 + threadIdx.x;
    if (i < n) p[i] = 0.0f;
}

__global__ void k_count_deg(const int* __restrict__ dst, int* __restrict__ cnt, int e) {
    int i = blockIdx.x * 256 + threadIdx.x;
    if (i < e) atomicAdd(&cnt[dst[i]], 1);
}

__global__ void k_dinv(const int* __restrict__ cnt, float* __restrict__ dinv, int n) {
    int i = blockIdx.x * 256 + threadIdx.x;
    if (i < n) dinv[i] = rsqrtf((float)(cnt[i] + 1));   // +1 self-loop; deg >= 1 always
}

// hagg = bias + self-loop contribution (dinv^2 * hlin), non-atomic init
__global__ void k_agg_init(const float* __restrict__ hlin, const float* __restrict__ dinv,
                           const float* __restrict__ bias, float* __restrict__ hagg, int n)
{
    int i = blockIdx.x * 256 + threadIdx.x;
    if (i >= n * 128) return;
    int node = i >> 7, c = i & 127;
    float di = dinv[node];
    hagg[i] = bias[c] + hlin[i] * di * di;
}

// one wave per edge: 32 lanes x float4 = 128 features, scatter with float atomics
__global__ void k_scatter(const int* __restrict__ src, const int* __restrict__ dst,
                          const float* __restrict__ dinv, const float* __restrict__ hlin,
                          float* __restrict__ hagg, int e)
{
    int t = blockIdx.x * 256 + threadIdx.x;
    int edge = t >> 5;
    if (edge >= e) return;
    int lane = t & 31;
    int s = src[edge], d = dst[edge];
    float nm = dinv[s] * dinv[d];
    float4 v = ((const float4*)(hlin + (size_t)s * 128))[lane];
    float* o = hagg + (size_t)d * 128 + (lane << 2);
    atomicAdd(o + 0, v.x * nm);
    atomicAdd(o + 1, v.y * nm);
    atomicAdd(o + 2, v.z * nm);
    atomicAdd(o + 3, v.w * nm);
}

// per-feature sum / sumsq; blockDim = 128 (one feature per thread, coalesced)
__global__ void k_gn_stats(const float* __restrict__ h, float* __restrict__ stats, int n)
{
    int c = threadIdx.x;
    float s = 0.0f, s2 = 0.0f;
    for (int node = blockIdx.x; node < n; node += gridDim.x) {
        float v = h[(size_t)node * 128 + c];
        s += v; s2 += v * v;
    }
    atomicAdd(&stats[c], s);
    atomicAdd(&stats[128 + c], s2);
}

// GraphNorm + ReLU: var(h - a*m) = E[h^2] - m^2 * a * (2 - a)
__global__ void k_gnorm_relu(const float* __restrict__ h, const float* __restrict__ stats,
                             const float* __restrict__ w, const float* __restrict__ b,
                             const float* __restrict__ a, float* __restrict__ out,
                             int n, float invN)
{
    int i = blockIdx.x * 256 + threadIdx.x;
    if (i >= n * 128) return;
    int c = i & 127;
    float m   = stats[c] * invN;
    float al  = a[c];
    float var = stats[128 + c] * invN - m * m * al * (2.0f - al);
    float cv  = h[i] - al * m;
    float y   = cv * rsqrtf(var + 1e-5f) * w[c] + b[c];
    out[i] = fmaxf(y, 0.0f);
}

// final projection [128 -> 2]: one wave per node, shuffle reduction
__global__ void k_out(const float* __restrict__ h, const float* __restrict__ Wo,
                      const float* __restrict__ bo, float* __restrict__ out, int n)
{
    int t = blockIdx.x * 256 + threadIdx.x;
    int node = t >> 5;
    if (node >= n) return;
    int lane = t & 31;
    float4 v = ((const float4*)(h + (size_t)node * 128))[lane];
    int k = lane << 2;
    float p0 = v.x * Wo[2*k+0] + v.y * Wo[2*k+2] + v.z * Wo[2*k+4] + v.w * Wo[2*k+6];
    float p1 = v.x * Wo[2*k+1] + v.y * Wo[2*k+3] + v.z * Wo[2*k+5] + v.w * Wo[2*k+7];
#pragma unroll
    for (int off = 16; off > 0; off >>= 1) {
        p0 += __shfl_xor(p0, off, 32);
        p1 += __shfl_xor(p1, off, 32);
    }
    if (lane == 0) {
        out[node * 2 + 0] = p0 + bo[0];
        out[node * 2 + 1] = p1 + bo[1];
    }
}

// ---------------------------------------------------------------------------
extern "C" void kernel_launch(void* const* d_in, const int* in_sizes, int n_in,
                              void* d_out, int out_size, void* d_ws, size_t ws_size,
                              hipStream_t stream)
{
    const float* x   = (const float*)d_in[0];
    const int*   ei  = (const int*)  d_in[1];
    const float* W1  = (const float*)d_in[2];
    const float* b1  = (const float*)d_in[3];
    const float* W2  = (const float*)d_in[4];
    const float* b2  = (const float*)d_in[5];
    const float* W3  = (const float*)d_in[6];
    const float* b3  = (const float*)d_in[7];
    const float* g1w = (const float*)d_in[8];
    const float* g1b = (const float*)d_in[9];
    const float* g1a = (const float*)d_in[10];
    const float* g2w = (const float*)d_in[11];
    const float* g2b = (const float*)d_in[12];
    const float* g2a = (const float*)d_in[13];
    const float* Wo  = (const float*)d_in[14];
    const float* bo  = (const float*)d_in[15];
    float* out = (float*)d_out;

    const int n = in_sizes[0] / 128;
    const int e = in_sizes[1] / 2;
    const int* src = ei;
    const int* dst = ei + e;

    const size_t N128 = (size_t)n * 128;
    float* ws    = (float*)d_ws;
    float* hIn   = ws;
    float* hLin  = ws + N128;
    float* hAgg  = ws + 2 * N128;
    float* dinv  = ws + 3 * N128;
    int*   cnt   = (int*)(dinv + n);
    float* stats = (float*)(cnt + n);

    dim3 blk(256);
    const int gElem = (int)((N128 + 255) / 256);
    const int gNode = (n + 255) / 256;
    const int gEdge = (e + 255) / 256;
    const int gScat = (int)(((size_t)e * 32 + 255) / 256);
    const int gGemm = (n + 127) / 128;
    const int gOut  = (int)(((size_t)n * 32 + 255) / 256);
    const float invN = 1.0f / (float)n;

    // symmetric-normalization degrees (same for all layers)
    k_zero_i32<<<gNode, blk, 0, stream>>>(cnt, n);
    k_count_deg<<<gEdge, blk, 0, stream>>>(dst, cnt, e);
    k_dinv<<<gNode, blk, 0, stream>>>(cnt, dinv, n);

    // ---- layer 1 ----
    k_gemm128<<<gGemm, blk, 0, stream>>>(x, W1, hLin, n);
    k_agg_init<<<gElem, blk, 0, stream>>>(hLin, dinv, b1, hAgg, n);
    k_scatter<<<gScat, blk, 0, stream>>>(src, dst, dinv, hLin, hAgg, e);
    k_zero_f32<<<1, blk, 0, stream>>>(stats, 256);
    k_gn_stats<<<512, dim3(128), 0, stream>>>(hAgg, stats, n);
    k_gnorm_relu<<<gElem, blk, 0, stream>>>(hAgg, stats, g1w, g1b, g1a, hIn, n, invN);

    // ---- layer 2 ----
    k_gemm128<<<gGemm, blk, 0, stream>>>(hIn, W2, hLin, n);
    k_agg_init<<<gElem, blk, 0, stream>>>(hLin, dinv, b2, hAgg, n);
    k_scatter<<<gScat, blk, 0, stream>>>(src, dst, dinv, hLin, hAgg, e);
    k_zero_f32<<<1, blk, 0, stream>>>(stats, 256);
    k_gn_stats<<<512, dim3(128), 0, stream>>>(hAgg, stats, n);
    k_gnorm_relu<<<gElem, blk, 0, stream>>>(hAgg, stats, g2w, g2b, g2a, hIn, n, invN);

    // ---- layer 3 + output head ----
    k_gemm128<<<gGemm, blk, 0, stream>>>(hIn, W3, hLin, n);
    k_agg_init<<<gElem, blk, 0, stream>>>(hLin, dinv, b3, hAgg, n);
    k_scatter<<<gScat, blk, 0, stream>>>(src, dst, dinv, hLin, hAgg, e);
    k_out<<<gOut, blk, 0, stream>>>(hAgg, Wo, bo, out, n);
}